// GNNModel_15195594293395
// MI455X (gfx1250) — compile-verified
//
#include <hip/hip_runtime.h>

typedef __attribute__((ext_vector_type(2))) float v2f;
typedef __attribute__((ext_vector_type(8))) float v8f;

#define WMMA_F32(a, b, c) \
    __builtin_amdgcn_wmma_f32_16x16x4_f32(false, (a), false, (b), (short)0, (c), false, false)

// ---------------------------------------------------------------------------
// Fused GEMM: C = act(A[M,K] @ B[K,N] + bias)
// Requirements (guaranteed by all call sites): M % 128 == 0, N % 64 == 0,
// lda even (8B-aligned v2f row loads), ldb % 4 == 0 (16B float4 loads).
// 128 threads = 4 waves; block tile 128x64; wave tile 32x64.
// K handled in 16-wide chunks staged in LDS; K-tail zero-filled in the
// loaders so the WMMA loop is branch-free.
// ---------------------------------------------------------------------------
__global__ __launch_bounds__(128)
void gemm_f32_wmma(const float* __restrict__ A, int lda,
                   const float* __restrict__ B, int ldb,
                   const float* __restrict__ bias,
                   float* __restrict__ C, int ldc,
                   int M, int N, int K, int relu)
{
    __shared__ float As[128][18];   // [row][k]  (pad 18: keeps 8B alignment, skews banks)
    __shared__ float Bs[64][18];    // [col][k]  transposed chunk -> ds_load_b64 per operand

    const int tid   = threadIdx.x;
    const int lane  = tid & 31;
    const int wv    = tid >> 5;
    const int mr    = lane & 15;            // A-row / B-col / C-col within 16-tile
    const int khalf = (lane >> 4) << 1;     // 0 or 2
    const int rbase = (lane >> 4) << 3;     // C rows 0..7 or 8..15
    const int mblk  = blockIdx.y * 128;
    const int n0    = blockIdx.x * 64;

    v8f zero = {};
    v8f acc[2][4];
#pragma unroll
    for (int i = 0; i < 2; ++i)
#pragma unroll
        for (int j = 0; j < 4; ++j) acc[i][j] = zero;

    for (int kc = 0; kc < K; kc += 16) {
        // ---- stage A chunk: thread t owns row (mblk+t), 16 k-values ----
        {
            const float* ap = A + (size_t)(mblk + tid) * lda + kc;
#pragma unroll
            for (int c = 0; c < 16; c += 2) {
                v2f v = {};
                if (kc + c < K) v = *(const v2f*)(ap + c);   // lda even, kc+c even -> 8B aligned
                As[tid][c]     = v.x;
                As[tid][c + 1] = v.y;
            }
        }
        // ---- stage B chunk transposed: 16 rows x 64 cols, float4 loads ----
        {
            const int r  = tid >> 3;            // 0..15
            const int cb = (tid & 7) * 8;       // 0..56
            float4 v0 = {}, v1 = {};
            if (kc + r < K) {
                const float* bp = B + (size_t)(kc + r) * ldb + n0 + cb;
                v0 = *(const float4*)bp;
                v1 = *(const float4*)(bp + 4);
            }
            Bs[cb + 0][r] = v0.x; Bs[cb + 1][r] = v0.y;
            Bs[cb + 2][r] = v0.z; Bs[cb + 3][r] = v0.w;
            Bs[cb + 4][r] = v1.x; Bs[cb + 5][r] = v1.y;
            Bs[cb + 6][r] = v1.z; Bs[cb + 7][r] = v1.w;
        }
        __syncthreads();

        // ---- branch-free WMMA loop over the chunk ----
#pragma unroll
        for (int kk = 0; kk < 16; kk += 4) {
            const int kp = kk + khalf;
            v2f a0 = *(const v2f*)&As[wv * 32 + mr][kp];
            v2f a1 = *(const v2f*)&As[wv * 32 + 16 + mr][kp];
#pragma unroll
            for (int nt = 0; nt < 4; ++nt) {
                v2f bb = *(const v2f*)&Bs[nt * 16 + mr][kp];
                acc[0][nt] = WMMA_F32(a0, bb, acc[0][nt]);
                acc[1][nt] = WMMA_F32(a1, bb, acc[1][nt]);
            }
        }
        __syncthreads();
    }

    // ---- epilogue: bias + activation, no bounds checks needed ----
#pragma unroll
    for (int mt = 0; mt < 2; ++mt)
#pragma unroll
        for (int nt = 0; nt < 4; ++nt) {
            const int col = n0 + nt * 16 + mr;
            const float bv = bias ? bias[col] : 0.0f;
#pragma unroll
            for (int r = 0; r < 8; ++r) {
                const int row = mblk + wv * 32 + mt * 16 + rbase + r;
                float v = acc[mt][nt][r] + bv;
                if (relu) v = fmaxf(v, 0.0f);
                C[(size_t)row * ldc + col] = v;
            }
        }
}

// ---------------------------------------------------------------------------
// Edge MLP, layer-2 fused with the factored layer-1 gather:
//   hid[e,c] = ReLU(P[send(e),c] + Q[rec(e),c] + b1[c])  (LDS resident, 16x256)
//   edge_out[e,:] = ReLU(hid @ W2[256,128] + b2)
// One block = 16 edges of one batch; wave w covers output cols [32w, 32w+32).
// W2 staged per 16-k chunk, transposed, via float4 loads (no redundancy:
// the 4 waves cover disjoint 32-col ranges, loader covers all 128 cols once).
// ---------------------------------------------------------------------------
__global__ __launch_bounds__(128)
void edge_mlp_kernel(const float* __restrict__ P, const float* __restrict__ Q,
                     const float* __restrict__ b1,
                     const float* __restrict__ W2, const float* __restrict__ b2,
                     float* __restrict__ edge_out)
{
    __shared__ float hid[16][258];   // row stride 258 floats (8B-aligned, skewed)
    __shared__ float W2T[128][18];   // [col][k] chunk

    const int tid  = threadIdx.x;
    const int b    = blockIdx.x / 62;
    const int tile = blockIdx.x % 62;
    const int e0   = tile * 16;

    // phase 1: build hid = ReLU(P[send] + Q[rec] + b1)
    for (int idx = tid; idx < 16 * 256; idx += 128) {
        const int r   = idx >> 8;
        const int c   = idx & 255;
        const int e   = e0 + r;
        const int s   = e / 31;
        const int cc  = e - s * 31;
        const int rec = cc + (cc >= s ? 1 : 0);
        const float v = P[((size_t)(b * 32 + s))   * 256 + c]
                      + Q[((size_t)(b * 32 + rec)) * 256 + c] + b1[c];
        hid[r][c] = fmaxf(v, 0.0f);
    }
    __syncthreads();

    const int lane  = tid & 31;
    const int wv    = tid >> 5;
    const int mr    = lane & 15;
    const int khalf = (lane >> 4) << 1;
    const int rbase = (lane >> 4) << 3;
    const int c0    = wv * 32;

    v8f zero = {};
    v8f acc[2]; acc[0] = zero; acc[1] = zero;

    for (int kc = 0; kc < 256; kc += 16) {
        // stage W2 chunk transposed: rows kc..kc+15, all 128 cols
        {
            const int r  = tid >> 3;             // 0..15
            const int cb = (tid & 7) * 16;       // 0..112
            const float* wp = W2 + (size_t)(kc + r) * 128 + cb;   // 16B aligned
            float4 v0 = *(const float4*)(wp + 0);
            float4 v1 = *(const float4*)(wp + 4);
            float4 v2 = *(const float4*)(wp + 8);
            float4 v3 = *(const float4*)(wp + 12);
            W2T[cb + 0][r] = v0.x;  W2T[cb + 1][r] = v0.y;
            W2T[cb + 2][r] = v0.z;  W2T[cb + 3][r] = v0.w;
            W2T[cb + 4][r] = v1.x;  W2T[cb + 5][r] = v1.y;
            W2T[cb + 6][r] = v1.z;  W2T[cb + 7][r] = v1.w;
            W2T[cb + 8][r] = v2.x;  W2T[cb + 9][r] = v2.y;
            W2T[cb + 10][r] = v2.z; W2T[cb + 11][r] = v2.w;
            W2T[cb + 12][r] = v3.x; W2T[cb + 13][r] = v3.y;
            W2T[cb + 14][r] = v3.z; W2T[cb + 15][r] = v3.w;
        }
        __syncthreads();

#pragma unroll
        for (int kk = 0; kk < 16; kk += 4) {
            const int kp = kk + khalf;
            v2f a = *(const v2f*)&hid[mr][kc + kp];
#pragma unroll
            for (int nt = 0; nt < 2; ++nt) {
                v2f bb = *(const v2f*)&W2T[c0 + nt * 16 + mr][kp];
                acc[nt] = WMMA_F32(a, bb, acc[nt]);
            }
        }
        __syncthreads();
    }

#pragma unroll
    for (int nt = 0; nt < 2; ++nt) {
        const int col = c0 + nt * 16 + mr;
        const float bv = b2[col];
#pragma unroll
        for (int r = 0; r < 8; ++r) {
            const int e = e0 + rbase + r;
            edge_out[((size_t)(b * 992 + e)) * 128 + col] = fmaxf(acc[nt][r] + bv, 0.0f);
        }
    }
}

// ---------------------------------------------------------------------------
// edge2node: msg[b,n,d] = mean over 31 incoming edges.
// Incoming edge from s to n: e = s*31 + (n < s ? n : n-1).
// ---------------------------------------------------------------------------
__global__ __launch_bounds__(128)
void edge2node_kernel(const float* __restrict__ edge_out, float* __restrict__ msg)
{
    const int b = blockIdx.x >> 5;
    const int n = blockIdx.x & 31;
    const int d = threadIdx.x;
    float s = 0.0f;
#pragma unroll
    for (int src = 0; src < 32; ++src) {
        if (src == n) continue;
        const int cc = (n < src) ? n : n - 1;
        const int e  = src * 31 + cc;
        s += edge_out[((size_t)(b * 992 + e)) * 128 + d];
    }
    msg[((size_t)(b * 32 + n)) * 128 + d] = s * (1.0f / 31.0f);
}

// ni = concat(msg[512,128], dec_t[512,3], zero-pad) -> [512,132] (even lda)
__global__ void build_ni_kernel(const float* __restrict__ msg,
                                const float* __restrict__ dec_t,
                                float* __restrict__ ni)
{
    int idx = blockIdx.x * blockDim.x + threadIdx.x;
    if (idx >= 512 * 132) return;
    int row = idx / 132, j = idx - row * 132;
    float v;
    if (j < 128)      v = msg[row * 128 + j];
    else if (j < 131) v = dec_t[row * 3 + (j - 128)];
    else              v = 0.0f;
    ni[idx] = v;
}

// GRU gate pointwise: Gih/Ghh hold [r|i|n] 128-col blocks (ldc=384).
__global__ void gru_kernel(const float* __restrict__ Gih, const float* __restrict__ Ghh,
                           const float* __restrict__ h, float* __restrict__ hout,
                           float* __restrict__ out_t)
{
    int idx = blockIdx.x * blockDim.x + threadIdx.x;
    if (idx >= 512 * 128) return;
    int row = idx >> 7, d = idx & 127;
    const float* gi = Gih + (size_t)row * 384;
    const float* gh = Ghh + (size_t)row * 384;
    float r  = 1.0f / (1.0f + __expf(-(gi[d]       + gh[d])));
    float i  = 1.0f / (1.0f + __expf(-(gi[128 + d] + gh[128 + d])));
    float nn = tanhf(gi[256 + d] + r * gh[256 + d]);
    float hv = h[idx];
    float hnew = (1.0f - i) * nn + i * hv;
    hout[idx]  = hnew;
    out_t[idx] = hnew;
}

// ---------------------------------------------------------------------------
extern "C" void kernel_launch(void* const* d_in, const int* in_sizes, int n_in,
                              void* d_out, int out_size, void* d_ws, size_t ws_size,
                              hipStream_t stream)
{
    const float* enc_in = (const float*)d_in[0];
    const float* dec_in = (const float*)d_in[1];
    // d_in[2]=R, d_in[3]=S: incidence handled analytically, not needed.
    const float* enc_w1 = (const float*)d_in[4];
    const float* enc_b1 = (const float*)d_in[5];
    const float* enc_w2 = (const float*)d_in[6];
    const float* enc_b2 = (const float*)d_in[7];
    const float* pe_w1  = (const float*)d_in[8];
    const float* pe_b1  = (const float*)d_in[9];
    const float* pe_w2  = (const float*)d_in[10];
    const float* pe_b2  = (const float*)d_in[11];
    const float* pn_w1  = (const float*)d_in[12];
    const float* pn_b1  = (const float*)d_in[13];
    const float* pn_w2  = (const float*)d_in[14];
    const float* pn_b2  = (const float*)d_in[15];
    const float* de_w1  = (const float*)d_in[16];
    const float* de_b1  = (const float*)d_in[17];
    const float* de_w2  = (const float*)d_in[18];
    const float* de_b2  = (const float*)d_in[19];
    const float* w_hr   = (const float*)d_in[20];
    const float* w_hi   = (const float*)d_in[21];
    const float* w_hn   = (const float*)d_in[22];
    const float* w_ir   = (const float*)d_in[23];
    const float* b_ir   = (const float*)d_in[24];
    const float* w_ii   = (const float*)d_in[25];
    const float* b_ii   = (const float*)d_in[26];
    const float* w_in   = (const float*)d_in[27];
    const float* b_in   = (const float*)d_in[28];

    float* out = (float*)d_out;

    // Workspace layout (floats); total ~12.3 MB -> resident in 192 MB L2.
    float* ws   = (float*)d_ws;
    float* h0   = ws;                 // 512*128
    float* h1   = h0  + 512 * 128;
    float* Pb   = h1  + 512 * 128;    // 512*256
    float* Qb   = Pb  + 512 * 256;
    float* tmp  = Qb  + 512 * 256;    // 512*256 (MLP hidden)
    float* eout = tmp + 512 * 256;    // 16*992*128
    float* msg  = eout + (size_t)16 * 992 * 128;
    float* ni   = msg + 512 * 128;    // 512*132 (padded)
    float* Gih  = ni  + 512 * 132;    // 512*384
    float* Ghh  = Gih + 512 * 384;

    auto gemm = [&](const float* A, int lda, const float* B, int ldb,
                    const float* bias, float* C, int ldc,
                    int M, int N, int K, int relu) {
        dim3 grid((N + 63) / 64, (M + 127) / 128);
        gemm_f32_wmma<<<grid, 128, 0, stream>>>(A, lda, B, ldb, bias, C, ldc, M, N, K, relu);
    };

    // ---- encoder MLP: [512,150] -> 256 -> 128 ----
    gemm(enc_in, 150, enc_w1, 256, enc_b1, tmp, 256, 512, 256, 150, 1);
    gemm(tmp,    256, enc_w2, 128, enc_b2, h0,  128, 512, 128, 256, 1);

    // ---- 2 message-passing rounds (factored node2edge) ----
    for (int pass = 0; pass < 2; ++pass) {
        gemm(h0, 128, pe_w1,             256, nullptr, Pb, 256, 512, 256, 128, 0); // x@W1a
        gemm(h0, 128, pe_w1 + 128 * 256, 256, nullptr, Qb, 256, 512, 256, 128, 0); // x@W1b
        edge_mlp_kernel<<<dim3(16 * 62), 128, 0, stream>>>(Pb, Qb, pe_b1, pe_w2, pe_b2, eout);
        edge2node_kernel<<<dim3(16 * 32), 128, 0, stream>>>(eout, msg);
        gemm(msg, 128, pn_w1, 256, pn_b1, tmp, 256, 512, 256, 128, 1);
        gemm(tmp, 256, pn_w2, 128, pn_b2, h0,  128, 512, 128, 256, 1);
    }

    // ---- decoder: 25 GRU message-passing steps ----
    float* hcur = h0;
    float* hnxt = h1;
    for (int t = 0; t < 25; ++t) {
        gemm(hcur, 128, de_w1,             256, nullptr, Pb, 256, 512, 256, 128, 0);
        gemm(hcur, 128, de_w1 + 128 * 256, 256, nullptr, Qb, 256, 512, 256, 128, 0);
        edge_mlp_kernel<<<dim3(16 * 62), 128, 0, stream>>>(Pb, Qb, de_b1, de_w2, de_b2, eout);
        edge2node_kernel<<<dim3(16 * 32), 128, 0, stream>>>(eout, msg);

        const float* dec_t = dec_in + (size_t)t * 512 * 3;
        build_ni_kernel<<<(512 * 132 + 255) / 256, 256, 0, stream>>>(msg, dec_t, ni);

        // input-side gates: [512,131] @ [131,128] + b  (ni padded to lda=132)
        gemm(ni, 132, w_ir, 128, b_ir, Gih + 0,   384, 512, 128, 131, 0);
        gemm(ni, 132, w_ii, 128, b_ii, Gih + 128, 384, 512, 128, 131, 0);
        gemm(ni, 132, w_in, 128, b_in, Gih + 256, 384, 512, 128, 131, 0);
        // hidden-side gates: [512,128] @ [128,128]
        gemm(hcur, 128, w_hr, 128, nullptr, Ghh + 0,   384, 512, 128, 128, 0);
        gemm(hcur, 128, w_hi, 128, nullptr, Ghh + 128, 384, 512, 128, 128, 0);
        gemm(hcur, 128, w_hn, 128, nullptr, Ghh + 256, 384, 512, 128, 128, 0);

        float* out_t = out + (size_t)t * 512 * 128;
        gru_kernel<<<(512 * 128 + 255) / 256, 256, 0, stream>>>(Gih, Ghh, hcur, hnxt, out_t);

        float* sw = hcur; hcur = hnxt; hnxt = sw;
    }

    // final state appended after outputs: hcur holds h after step 24
    hipMemcpyAsync(out + (size_t)25 * 512 * 128, hcur, (size_t)512 * 128 * sizeof(float),
                   hipMemcpyDeviceToDevice, stream);
}